// MatchaTTS_67362267071009
// MI455X (gfx1250) — compile-verified
//
#include <hip/hip_runtime.h>
#include <hip/hip_bf16.h>
#include <stdint.h>

#define BB 16
#define SS 512
#define TT 4096
#define CC 512

typedef float f32x4 __attribute__((ext_vector_type(4)));
typedef int   v4i   __attribute__((ext_vector_type(4)));
#define GLOBAL_AS __attribute__((address_space(1)))
#define SHARED_AS __attribute__((address_space(3)))

#if defined(__has_builtin)
#if __has_builtin(__builtin_amdgcn_global_load_async_to_lds_b128) && \
    __has_builtin(__builtin_amdgcn_global_store_async_from_lds_b128) && \
    __has_builtin(__builtin_amdgcn_s_wait_asynccnt)
#define USE_ASYNC 1
#endif
#endif

// ---------------------------------------------------------------------------
// Kernel 2 (defined first so its device asm leads the disassembly):
// one 128-thread block per output row (B*T = 65536 blocks). Row copy =
// 512 floats. enc_out (16 MB) stays hot in the 192 MB L2; the 128 MB output
// streams out via the gfx1250 async Global->LDS->Global DMA path (ASYNCcnt,
// no VGPR staging). Row condition is uniform per block -> EXEC all-ones
// inside the async branch, as the ISA requires.
// ---------------------------------------------------------------------------
__global__ void regulate_gather(const float* __restrict__ enc,
                                const int* __restrict__ map,
                                float* __restrict__ out) {
  __shared__ float stage[CC];
  const int bt  = blockIdx.x;
  const int b   = bt >> 12;            // T = 4096
  const int tid = threadIdx.x;         // 128 threads, 16B each

  const int s = map[bt];               // uniform per block
  f32x4* dst = (f32x4*)(out + ((size_t)bt << 9));

  if (s >= 0) {
    const f32x4* src = (const f32x4*)(enc + ((size_t)(b * SS + s) << 9));
#if USE_ASYNC
    // Global -> LDS (async DMA), wait ASYNCcnt==0 (LDS side of async ops is
    // unordered, so the wait is required), then LDS -> Global (async DMA).
    __builtin_amdgcn_global_load_async_to_lds_b128(
        (GLOBAL_AS v4i*)(void*)(src + tid),
        (SHARED_AS v4i*)(stage + tid * 4), 0, 0);
    __builtin_amdgcn_s_wait_asynccnt(0);
    __builtin_amdgcn_global_store_async_from_lds_b128(
        (GLOBAL_AS v4i*)(void*)(dst + tid),
        (SHARED_AS v4i*)(stage + tid * 4), 0, 0);
    // s_endpgm performs an implicit wait-idle, draining ASYNCcnt.
#else
    f32x4 v = src[tid];
    __builtin_nontemporal_store(v, dst + tid);
#endif
  } else {
    f32x4 z = {0.f, 0.f, 0.f, 0.f};
    __builtin_nontemporal_store(z, dst + tid);
  }
}

// ---------------------------------------------------------------------------
// Kernel 1: one block per batch. Computes reps (== durations for pace=1),
// inclusive scan over S (Hillis-Steele in LDS), writes dec_lens and reps to
// the output tail, and builds the t -> s segment map in workspace.
// ---------------------------------------------------------------------------
__global__ void regulate_setup(const int* __restrict__ dur,
                               float* __restrict__ out,
                               int* __restrict__ map) {
  __shared__ int tmp[SS];
  const int b = blockIdx.x;
  const int s = threadIdx.x;           // 512 threads

  const int d0 = dur[b * SS + s];
  const int d  = (int)((float)d0 / 1.0f + 0.5f);   // mirrors reference rounding

  // reps output (float32), after enc_rep and dec_lens in the flat buffer
  out[(size_t)BB * TT * CC + BB + (size_t)b * SS + s] = (float)d;

  tmp[s] = d;
  __syncthreads();
  for (int off = 1; off < SS; off <<= 1) {
    int v = (s >= off) ? tmp[s - off] : 0;
    __syncthreads();
    tmp[s] += v;
    __syncthreads();
  }
  const int incl = tmp[s];
  const int excl = incl - d;

  if (s == SS - 1) {
    int dl = incl < TT ? incl : TT;
    out[(size_t)BB * TT * CC + b] = (float)dl;     // dec_lens
  }

  int* mrow = map + (size_t)b * TT;
  for (int t = s; t < TT; t += SS) mrow[t] = -1;   // rows past dec_len -> zeros
  __syncthreads();                                  // barrier + block mem fence

  const int end = incl < TT ? incl : TT;
  for (int t = excl; t < end; ++t) mrow[t] = s;    // disjoint segments, 1 writer/t
}

extern "C" void kernel_launch(void* const* d_in, const int* in_sizes, int n_in,
                              void* d_out, int out_size, void* d_ws, size_t ws_size,
                              hipStream_t stream) {
  const int*   durations = (const int*)d_in[0];     // [B,S] int32
  const float* enc_out   = (const float*)d_in[1];   // [B,S,C] f32
  (void)in_sizes; (void)n_in; (void)out_size; (void)ws_size;
  // d_in[2] is mel_max_len (scalar) -- static T=4096 here.

  float* out = (float*)d_out;          // [B*T*C] enc_rep | [B] dec_lens | [B*S] reps
  int*   map = (int*)d_ws;             // [B*T] segment map (256 KB)

  regulate_setup<<<dim3(BB), dim3(SS), 0, stream>>>(durations, out, map);
  regulate_gather<<<dim3(BB * TT), dim3(CC / 4), 0, stream>>>(enc_out, map, out);
}